// SchrodingerOperator_63909113365206
// MI455X (gfx1250) — compile-verified
//
#include <hip/hip_runtime.h>

#define NPTS 8192
#define NM   12

typedef __attribute__((ext_vector_type(2))) float v2f;
typedef __attribute__((ext_vector_type(4))) float v4f;
typedef __attribute__((ext_vector_type(8))) float v8f;

static __device__ __forceinline__ float coeff_at(const float* __restrict__ cc,
                                                 const float* __restrict__ sc, int k) {
  return (k < NM) ? cc[k] : sc[k - NM];
}

static __device__ __forceinline__ float basis_at(int k, float x) {
  const float twopi = 6.28318530717958647692f;
  if (k < NM) return cosf(twopi * (float)(k + 1) * x);
  return sinf(twopi * (float)(k - NM + 1) * x);
}

// ---------------------------------------------------------------------------
// V[0..8191] via V_WMMA_F32_16X16X4_F32.
// D = sum_s A_s x B_s, A[m][k] = coeff(4s+k) (rows replicated),
// B[k][n] = basis(4s+k, x_{tile*16+n}).  Every row of D equals V at 16 points.
// A 16x4 f32 layout: lane holds M=lane&15, K = (lane>=16?2:0)+vgpr.
// B 4x16 f32 layout: lane holds N=lane&15, K = (lane>=16?2:0)+vgpr.
// D 16x16 f32: vgpr0, lanes 0-15 => (M=0, N=lane).
// ---------------------------------------------------------------------------
__global__ void potential_wmma(const float* __restrict__ a0,
                               const float* __restrict__ cc,
                               const float* __restrict__ sc,
                               float* __restrict__ V) {
  const int lane = threadIdx.x & 31;
  const int wave = (int)((blockIdx.x * blockDim.x + threadIdx.x) >> 5);  // tile id
  const int n    = lane & 15;
  const int xi   = wave * 16 + n;
  const float x  = (float)xi * (1.0f / (float)(NPTS - 1));  // linspace(0,1,N)
  const int khalf = (lane >> 4) << 1;  // 0 or 2

  v8f c = {};
#pragma unroll
  for (int s = 0; s < 6; ++s) {
    const int k0 = 4 * s + khalf;
    v2f a, b;
    a.x = coeff_at(cc, sc, k0);
    a.y = coeff_at(cc, sc, k0 + 1);
    b.x = basis_at(k0, x);
    b.y = basis_at(k0 + 1, x);
    // 8 args: (neg_a, A, neg_b, B, c_mod, C, reuse_a, reuse_b)
    c = __builtin_amdgcn_wmma_f32_16x16x4_f32(false, a, false, b,
                                              (short)0, c, false, false);
  }
  if (lane < 16) V[xi] = c[0] + a0[0];
}

// ---------------------------------------------------------------------------
// Fill H: tridiagonal band + zeros everywhere else.
// One v4f (b128) store per 4 elements; zeros go out non-temporal (output is
// 256MB > 192MB L2 and never re-read on device).
// ---------------------------------------------------------------------------
template <bool USE_V>
__global__ void fill_H(const float* __restrict__ V,
                       const float* __restrict__ a0,
                       const float* __restrict__ cc,
                       const float* __restrict__ sc,
                       float* __restrict__ out) {
  const float inv_dx2   = 67108864.0f;    // 8192^2
  const float diag_base = 134217728.0f;   // 2 * 8192^2
  constexpr int    ITERS = 8;
  constexpr size_t TOTQ  = (size_t)NPTS * NPTS / 4;   // float4 groups
  constexpr size_t TT    = TOTQ / ITERS;              // total threads

  const size_t tid = (size_t)blockIdx.x * blockDim.x + threadIdx.x;
  const v4f zero4 = {0.0f, 0.0f, 0.0f, 0.0f};

#pragma unroll
  for (int it = 0; it < ITERS; ++it) {
    const size_t q   = tid + (size_t)it * TT;
    const int    row = (int)(q >> 11);            // 2048 float4 per row
    const int    c4  = (int)(q & 2047) << 2;      // first column of group
    const unsigned d = (unsigned)(row - c4 + 1);  // band overlap iff d <= 5
    if (d <= 5u) {
      float vr;
      if constexpr (USE_V) {
        vr = V[row];
      } else {
        vr = a0[0];
        const float x = (float)row * (1.0f / (float)(NPTS - 1));
#pragma unroll
        for (int k = 0; k < NM; ++k) {
          const float ph = 6.28318530717958647692f * (float)(k + 1) * x;
          vr += cc[k] * cosf(ph) + sc[k] * sinf(ph);
        }
      }
      v4f v;
#pragma unroll
      for (int j = 0; j < 4; ++j) {
        const int col = c4 + j;
        float val = 0.0f;
        if (col == row)                           val = diag_base + vr;
        else if (col == row - 1 || col == row + 1) val = -inv_dx2;
        v[j] = val;
      }
      *(v4f*)(out + (q << 2)) = v;
    } else {
      __builtin_nontemporal_store(zero4, (v4f*)(out + (q << 2)));
    }
  }
}

extern "C" void kernel_launch(void* const* d_in, const int* in_sizes, int n_in,
                              void* d_out, int out_size, void* d_ws, size_t ws_size,
                              hipStream_t stream) {
  (void)in_sizes; (void)n_in; (void)out_size;
  const float* a0 = (const float*)d_in[0];
  const float* cc = (const float*)d_in[1];
  const float* sc = (const float*)d_in[2];
  float* out = (float*)d_out;
  float* V   = (float*)d_ws;

  constexpr size_t TOTQ  = (size_t)NPTS * NPTS / 4;  // 16,777,216
  constexpr int    ITERS = 8;
  constexpr size_t THREADS = TOTQ / ITERS;           // 2,097,152
  const dim3 block(256);
  const dim3 grid((unsigned)(THREADS / 256));        // 8192 blocks

  if (ws_size >= (size_t)NPTS * sizeof(float)) {
    // 512 waves, each producing 16 values of V on the matrix unit.
    potential_wmma<<<64, 256, 0, stream>>>(a0, cc, sc, V);
    fill_H<true><<<grid, block, 0, stream>>>(V, a0, cc, sc, out);
  } else {
    fill_H<false><<<grid, block, 0, stream>>>(nullptr, a0, cc, sc, out);
  }
}